// BlockDCT2d_12378095747840
// MI455X (gfx1250) — compile-verified
//
#include <hip/hip_runtime.h>

// Blocked 8x8 DCT-II: per block Y = D * X * D^T.
// x: (16, 32, 512, 512) f32 ; dct: (8,8) f32 ; out: (16, 32*64, 64, 64) f32.
//
// Strategy: Dblk = diag(D, D) (16x16). For a 16x16 tile (2x2 blocks):
//   T = Dblk * X          (four v_wmma_f32_16x16x4_f32, K chunks of 4)
//   Y = T * Dblk^T        (four more, A-chunks of T read from LDS)
// A-layout(Dblk) registers == B-layout(Dblk^T) registers, so one copy serves both.

typedef __attribute__((ext_vector_type(2))) float v2f;
typedef __attribute__((ext_vector_type(8))) float v8f;

#define LDS_STRIDE 18            // 16 + 2 pad: bank-conflict-free, 8B-aligned pairs
#define TILE_LDS   (16 * LDS_STRIDE)

__global__ __launch_bounds__(256)
void block_dct8_wmma_kernel(const float* __restrict__ x,
                            const float* __restrict__ dct,
                            float* __restrict__ out)
{
    __shared__ __align__(16) float lds[8 * TILE_LDS];   // 9216 B

    const int tid  = threadIdx.x;
    const int wave = tid >> 5;
    const int lane = tid & 31;
    const int half = lane >> 4;    // 0: lanes 0-15, 1: lanes 16-31
    const int n16  = lane & 15;

    const int bc = blockIdx.z;                    // fused B*C  (512)
    const int h0 = blockIdx.y * 16;               // tile-strip row in H
    const int w0 = blockIdx.x * 128 + wave * 16;  // this wave's tile col in W

    // ---- Dblk chunks. A-layout of 16x4 chunk c (f32):
    //   VGPR0: lanes0-15 -> K=4c+0, lanes16-31 -> K=4c+2 ; VGPR1: K=4c+1 / 4c+3
    // Same registers are the B-layout of Dblk^T for pass 2.
    v2f dk[4];
    {
        const int m = n16;
        #pragma unroll
        for (int c = 0; c < 4; ++c) {
            const int k0 = 4 * c + 2 * half;            // even; k0,k0+1 same 8-block
            const bool inblk = (m >> 3) == (k0 >> 3);   // block-diagonal
            dk[c].x = inblk ? dct[(m & 7) * 8 + (k0 & 7)]       : 0.0f;
            dk[c].y = inblk ? dct[(m & 7) * 8 + ((k0 + 1) & 7)] : 0.0f;
        }
    }

    // ---- Load X tile directly into B-layout chunks (rows are 64B coalesced
    //      per half-wave; tile rows contiguous in memory, stride W=512).
    const float* xt = x + ((size_t)bc * 512 + h0) * 512 + w0;
    v2f xb[4];
    #pragma unroll
    for (int c = 0; c < 4; ++c) {
        const int r0 = 4 * c + 2 * half;
        xb[c].x = xt[(size_t)r0 * 512 + n16];
        xb[c].y = xt[(size_t)(r0 + 1) * 512 + n16];
    }

    // ---- Pass 1: T = Dblk * X  (accumulate over 4 K-chunks)
    v8f acc = {};
    #pragma unroll
    for (int c = 0; c < 4; ++c)
        acc = __builtin_amdgcn_wmma_f32_16x16x4_f32(
                  false, dk[c], false, xb[c], (short)0, acc, false, false);

    // ---- T -> LDS (C/D layout: VGPR r, lane L holds T[r + 8*half][n16])
    float* tl = &lds[wave * TILE_LDS];
    #pragma unroll
    for (int r = 0; r < 8; ++r)
        tl[(r + 8 * half) * LDS_STRIDE + n16] = acc[r];

    // ---- Pass 2: Y = T * Dblk^T. A-chunks of T read from LDS (transposed
    //      access, conflict-free via stride 18; even index -> 8B ds_load_b64).
    v8f y = {};
    #pragma unroll
    for (int c = 0; c < 4; ++c) {
        v2f a = *(const v2f*)&tl[n16 * LDS_STRIDE + 4 * c + 2 * half];
        y = __builtin_amdgcn_wmma_f32_16x16x4_f32(
                false, a, false, dk[c], (short)0, y, false, false);
    }

    // ---- Restage Y in LDS (reuse this wave's region; same-wave LDS is in-order)
    #pragma unroll
    for (int r = 0; r < 8; ++r)
        tl[(r + 8 * half) * LDS_STRIDE + n16] = y[r];
    __syncthreads();

    // ---- Cooperative coalesced store. Workgroup owns: 64 planes (p = k*8+l),
    //      2 block-rows (hr), 16 block-cols (wc). 2048 elems, 8 per thread.
    //      Consecutive tid -> consecutive wb -> 64B contiguous chunks.
    const int hb0 = blockIdx.y * 2;
    const int wb0 = blockIdx.x * 16;
    float* ob = out + (size_t)bc * (64 * 64 * 64);
    #pragma unroll
    for (int i = 0; i < 8; ++i) {
        const int e    = i * 256 + tid;     // 0..2047
        const int p    = e >> 5;            // plane = k*8 + l
        const int hr   = (e >> 4) & 1;
        const int wc   = e & 15;
        const int tile = wc >> 1;           // which wave's 16x16 tile
        const int M    = hr * 8 + (p >> 3); // row in tile   (k = p>>3)
        const int N    = (wc & 1) * 8 + (p & 7); // col in tile (l = p&7)
        const float v  = lds[tile * TILE_LDS + M * LDS_STRIDE + N];
        ob[((size_t)p * 64 + (hb0 + hr)) * 64 + (wb0 + wc)] = v;
    }
}

extern "C" void kernel_launch(void* const* d_in, const int* in_sizes, int n_in,
                              void* d_out, int out_size, void* d_ws, size_t ws_size,
                              hipStream_t stream) {
    const float* x   = (const float*)d_in[0];   // (16, 32, 512, 512) f32
    const float* dct = (const float*)d_in[1];   // (8, 8) f32
    float* out = (float*)d_out;                 // (16, 2048, 64, 64) f32

    // W strips: 512/128 = 4 ; H strips: 512/16 = 32 ; B*C = 512
    dim3 grid(4, 32, 512);
    dim3 block(256);
    block_dct8_wmma_kernel<<<grid, block, 0, stream>>>(x, dct, out);
}